// MeanSpanRepr_69750268887123
// MI455X (gfx1250) — compile-verified
//
#include <hip/hip_runtime.h>
#include <hip/hip_bf16.h>

// Problem constants (match reference)
#define BSZ 8
#define SEQ 2048
#define HD  1024
#define PD  256
#define NQ  8192
#define MROWS (BSZ * SEQ)        // 16384
#define CHUNK 128
#define NCH  (SEQ / CHUNK)       // 16

typedef __attribute__((ext_vector_type(2))) float v2f;
typedef __attribute__((ext_vector_type(8))) float v8f;

// ---------------------------------------------------------------------------
// Kernel 1: projection GEMM  X[M][PD] = enc[M][HD] @ W[HD][PD] + b[PD]
// via V_WMMA_F32_16X16X4_F32 (exact f32 math, matches reference precision).
//
// Wave tile: 16 (M) x 64 (N), 4 f32 accumulators, A fragment reused 4x.
// Block = 256 threads = 8 waves, each wave an independent M-tile.
// grid.x = MROWS/16/8 = 128, grid.y = PD/64 = 4.
//
// ISA fragment layouts (CDNA5 ISA):
//  A (16x4 f32):  lanes 0-15 -> M=lane,  VGPR{0,1} = K{0,1}
//                 lanes 16-31 -> M=lane-16, VGPR{0,1} = K{2,3}
//  B (4x16 f32):  row striped across lanes within a VGPR, half-split like C/D:
//                 VGPR0 = rows {K, K+2}, VGPR1 = rows {K+1, K+3}, col = lane&15
//  C/D (16x16):   VGPR r, lanes 0-15 -> M=r, lanes 16-31 -> M=r+8, N = lane&15
// ---------------------------------------------------------------------------
__global__ __launch_bounds__(256) void proj_gemm_wmma(
    const float* __restrict__ A,     // [MROWS][HD]
    const float* __restrict__ W,     // [HD][PD]
    const float* __restrict__ bias,  // [PD]
    float* __restrict__ X)           // [MROWS][PD]
{
    const int lane    = threadIdx.x & 31;
    const int wave    = threadIdx.x >> 5;
    const int l15     = lane & 15;
    const int halfSel = lane >> 4;   // 0: lanes 0-15, 1: lanes 16-31

    const int mBase = (blockIdx.x * 8 + wave) * 16;
    const int nBase = blockIdx.y * 64;

    // A: this lane reads row (mBase + l15), K offsets (k + halfSel*2 + {0,1})
    const float* aRow = A + (size_t)(mBase + l15) * HD + halfSel * 2;

    v8f acc0 = {}, acc1 = {}, acc2 = {}, acc3 = {};

#pragma unroll 4
    for (int k = 0; k < HD; k += 4) {
        v2f a;
        a.x = aRow[k];
        a.y = aRow[k + 1];

        // B rows k+halfSel*2 (VGPR0.x) and k+halfSel*2+1 (VGPR0.y)
        const float* w0 = W + (size_t)(k + halfSel * 2) * PD + nBase + l15;
        const float* w1 = w0 + PD;

        v2f b0, b1, b2, b3;
        b0.x = w0[0];   b0.y = w1[0];
        b1.x = w0[16];  b1.y = w1[16];
        b2.x = w0[32];  b2.y = w1[32];
        b3.x = w0[48];  b3.y = w1[48];

        acc0 = __builtin_amdgcn_wmma_f32_16x16x4_f32(false, a, false, b0, (short)0, acc0, false, false);
        acc1 = __builtin_amdgcn_wmma_f32_16x16x4_f32(false, a, false, b1, (short)0, acc1, false, false);
        acc2 = __builtin_amdgcn_wmma_f32_16x16x4_f32(false, a, false, b2, (short)0, acc2, false, false);
        acc3 = __builtin_amdgcn_wmma_f32_16x16x4_f32(false, a, false, b3, (short)0, acc3, false, false);
    }

    const float bv0 = bias[nBase + l15];
    const float bv1 = bias[nBase + 16 + l15];
    const float bv2 = bias[nBase + 32 + l15];
    const float bv3 = bias[nBase + 48 + l15];

#pragma unroll
    for (int r = 0; r < 8; ++r) {
        const int row = mBase + r + 8 * halfSel;
        float* xr = X + (size_t)row * PD + nBase + l15;
        xr[0]  = acc0[r] + bv0;
        xr[16] = acc1[r] + bv1;
        xr[32] = acc2[r] + bv2;
        xr[48] = acc3[r] + bv3;
    }
}

// ---------------------------------------------------------------------------
// Kernel 2a: per-chunk partial sums  part[b][ch][d] = sum_{i<CHUNK} X[b][ch*CHUNK+i][d]
// 32768 threads, coalesced across d.
// ---------------------------------------------------------------------------
__global__ __launch_bounds__(256) void chunk_partials(
    const float* __restrict__ X, float* __restrict__ part)
{
    const int idx = blockIdx.x * blockDim.x + threadIdx.x;  // [0, BSZ*NCH*PD)
    const int d  = idx % PD;
    const int ch = (idx / PD) % NCH;
    const int b  = idx / (PD * NCH);
    const float* p = X + ((size_t)b * SEQ + (size_t)ch * CHUNK) * PD + d;
    float s = 0.0f;
#pragma unroll 8
    for (int i = 0; i < CHUNK; ++i) s += p[(size_t)i * PD];
    part[idx] = s;
}

// ---------------------------------------------------------------------------
// Kernel 2b: in-place exclusive scan of partials over the chunk axis.
// 2048 threads (one per (b,d)), 16-step serial loop.
// ---------------------------------------------------------------------------
__global__ __launch_bounds__(256) void scan_partials(float* __restrict__ part)
{
    const int idx = blockIdx.x * blockDim.x + threadIdx.x;  // [0, BSZ*PD)
    const int d = idx % PD;
    const int b = idx / PD;
    float run = 0.0f;
#pragma unroll
    for (int ch = 0; ch < NCH; ++ch) {
        const size_t o = ((size_t)b * NCH + ch) * PD + d;
        const float v = part[o];
        part[o] = run;
        run += v;
    }
}

// ---------------------------------------------------------------------------
// Kernel 2c: write csum[b][t][d] for t in [0, SEQ]; csum[b][0][d] = 0.
// Each thread handles one (b, chunk, d) and runs through its chunk.
// ---------------------------------------------------------------------------
__global__ __launch_bounds__(256) void write_csum(
    const float* __restrict__ X, const float* __restrict__ part,
    float* __restrict__ csum)
{
    const int idx = blockIdx.x * blockDim.x + threadIdx.x;  // [0, BSZ*NCH*PD)
    const int d  = idx % PD;
    const int ch = (idx / PD) % NCH;
    const int b  = idx / (PD * NCH);

    float run = part[idx];  // exclusive prefix of earlier chunks
    const float* xp = X + ((size_t)b * SEQ + (size_t)ch * CHUNK) * PD + d;
    float* cp = csum + ((size_t)b * (SEQ + 1) + (size_t)ch * CHUNK + 1) * PD + d;

    if (ch == 0) csum[(size_t)b * (SEQ + 1) * PD + d] = 0.0f;

#pragma unroll 8
    for (int i = 0; i < CHUNK; ++i) {
        run += xp[(size_t)i * PD];
        cp[(size_t)i * PD] = run;
    }
}

// ---------------------------------------------------------------------------
// Kernel 3: span gather.  One block per query; thread d computes both spans.
// out = [res1 (NQ*PD) | res2 (NQ*PD)]
// ---------------------------------------------------------------------------
__global__ __launch_bounds__(256) void span_gather(
    const float* __restrict__ csum,
    const int* __restrict__ s1, const int* __restrict__ e1,
    const int* __restrict__ qb,
    const int* __restrict__ s2, const int* __restrict__ e2,
    float* __restrict__ out)
{
    const int q = blockIdx.x;
    const int d = threadIdx.x;
    const int b = qb[q];
    const float* base = csum + (size_t)b * (SEQ + 1) * PD;

    {
        const int s = s1[q], e = e1[q];
        const float inv = 1.0f / (float)(e - s + 1);
        const float v = (base[(size_t)(e + 1) * PD + d] - base[(size_t)s * PD + d]) * inv;
        out[(size_t)q * PD + d] = v;
    }
    {
        const int s = s2[q], e = e2[q];
        const float inv = 1.0f / (float)(e - s + 1);
        const float v = (base[(size_t)(e + 1) * PD + d] - base[(size_t)s * PD + d]) * inv;
        out[(size_t)NQ * PD + (size_t)q * PD + d] = v;
    }
}

// ---------------------------------------------------------------------------
// Launch
// ---------------------------------------------------------------------------
extern "C" void kernel_launch(void* const* d_in, const int* in_sizes, int n_in,
                              void* d_out, int out_size, void* d_ws, size_t ws_size,
                              hipStream_t stream) {
    // setup_inputs() order:
    // 0 flag, 1 encoded_input, 2 start_ids_1, 3 end_ids_1, 4 query_batch_idx,
    // 5 start_ids_2, 6 end_ids_2, 7 W, 8 b
    const float* enc  = (const float*)d_in[1];
    const int*   s1   = (const int*)d_in[2];
    const int*   e1   = (const int*)d_in[3];
    const int*   qb   = (const int*)d_in[4];
    const int*   s2   = (const int*)d_in[5];
    const int*   e2   = (const int*)d_in[6];
    const float* W    = (const float*)d_in[7];
    const float* bias = (const float*)d_in[8];
    float*       out  = (float*)d_out;

    // Workspace partition
    char* ws = (char*)d_ws;
    float* X    = (float*)ws;                                         // MROWS*PD   = 16 MB
    float* part = (float*)(ws + (size_t)MROWS * PD * sizeof(float));  // BSZ*NCH*PD = 128 KB
    float* csum = (float*)(ws + (size_t)MROWS * PD * sizeof(float)
                              + (size_t)BSZ * NCH * PD * sizeof(float)); // BSZ*(SEQ+1)*PD ~= 16.8 MB

    // 1) GEMM + bias (WMMA f32)
    proj_gemm_wmma<<<dim3(MROWS / 16 / 8, PD / 64), 256, 0, stream>>>(enc, W, bias, X);

    // 2) blocked prefix sum along sequence
    chunk_partials<<<(BSZ * NCH * PD) / 256, 256, 0, stream>>>(X, part);
    scan_partials<<<(BSZ * PD) / 256, 256, 0, stream>>>(part);
    write_csum<<<(BSZ * NCH * PD) / 256, 256, 0, stream>>>(X, part, csum);

    // 3) span means
    span_gather<<<NQ, 256, 0, stream>>>(csum, s1, e1, qb, s2, e2, out);
}